// SpatialLoss_55121610277037
// MI455X (gfx1250) — compile-verified
//
#include <hip/hip_runtime.h>
#include <cstdint>

#define TILE_H   128
#define TILE_W   128
#define PLANE    (TILE_H * TILE_W)
#define NTHREADS 256

typedef float v2f __attribute__((ext_vector_type(2)));
typedef float v8f __attribute__((ext_vector_type(8)));
typedef int   v4i __attribute__((vector_size(16)));

// ---------------- CDNA5 wave32 all-reduce via V_WMMA_F32_16X16X4_F32 ----------
// A (16x4 f32): VGPR0 lanes0-15 = (M=lane,K=0), lanes16-31 = (M=lane-16,K=2);
// VGPR1 = 0 -> K=1,3 are zero.  B = ones(4x16).  D[m][n] = acc[m] + acc[m+16].
// Per-lane sum of D's 8 VGPRs gives S_low (lanes 0-15) / S_high (lanes 16-31).
// Second identical WMMA: D2[m][n] = S_low + S_high = total, in every lane.
__device__ inline float wave_allreduce(float acc) {
#if __has_builtin(__builtin_amdgcn_wmma_f32_16x16x4_f32)
  v2f a;    a[0] = acc;  a[1] = 0.0f;
  v2f ones; ones[0] = 1.0f; ones[1] = 1.0f;
  v8f c = {0.f, 0.f, 0.f, 0.f, 0.f, 0.f, 0.f, 0.f};
  v8f d = __builtin_amdgcn_wmma_f32_16x16x4_f32(
      false, a, false, ones, (short)0, c, false, false);
  float s = ((d[0] + d[1]) + (d[2] + d[3])) + ((d[4] + d[5]) + (d[6] + d[7]));
  v2f a2; a2[0] = s; a2[1] = 0.0f;
  v8f d2 = __builtin_amdgcn_wmma_f32_16x16x4_f32(
      false, a2, false, ones, (short)0, c, false, false);
  return d2[0];
#else
  for (int m = 16; m >= 1; m >>= 1) acc += __shfl_xor(acc, m, 32);
  return acc;
#endif
}

// ---------------- Kernel 1: one block per (t,c) plane -------------------------
__global__ __launch_bounds__(NTHREADS)
void spatial_loss_plane(const float* __restrict__ x, float* __restrict__ partials)
{
  __shared__ float tile[PLANE];          // 64 KB, whole 128x128 plane
  const int tid   = threadIdx.x;
  const int plane = blockIdx.x;
  const float* src = x + (size_t)plane * PLANE;

  // ---- Stage the plane global -> LDS --------------------------------------
#if __has_builtin(__builtin_amdgcn_global_load_async_to_lds_b128)
  {
    typedef __attribute__((address_space(1))) char gchar;
    typedef __attribute__((address_space(3))) char lchar;
    typedef __attribute__((address_space(1))) v4i  g_v4i;
    typedef __attribute__((address_space(3))) v4i  l_v4i;
    gchar* gbase = (gchar*)src;            // addrspacecast + const cast
    lchar* lbase = (lchar*)tile;
#pragma unroll
    for (int k = 0; k < PLANE / (4 * NTHREADS); ++k) {
      const int g = (tid + k * NTHREADS) * 16;   // byte offset, 16B aligned
      __builtin_amdgcn_global_load_async_to_lds_b128(
          (g_v4i*)(gbase + g), (l_v4i*)(lbase + g), 0, 0);
    }
  }
#if __has_builtin(__builtin_amdgcn_s_wait_asynccnt)
  __builtin_amdgcn_s_wait_asynccnt(0);
#else
  asm volatile("s_wait_asynccnt 0" ::: "memory");
#endif
  __syncthreads();
#else
  // Fallback: vectorized load + LDS store
  for (int k = 0; k < PLANE / (4 * NTHREADS); ++k) {
    const int g = tid + k * NTHREADS;
    ((float4*)tile)[g] = ((const float4*)src)[g];
  }
  __syncthreads();
#endif

  // ---- 12 symmetric offsets (i>0, or i==0 && j>0); factor 2 applied at end.
  // scale_k = 1 / ((H-|i|)*(W-|j|)*sqrt(i*i+j*j))  -- constexpr-folded.
  constexpr int   NOFF = 12;
  const int   oi[NOFF] = {0, 0, 1, 1, 1, 1, 1, 2, 2, 2, 2, 2};
  const int   oj[NOFF] = {1, 2,-2,-1, 0, 1, 2,-2,-1, 0, 1, 2};
  const float rd[NOFF] = {1.0f, 2.0f, 2.2360679775f, 1.41421356237f, 1.0f,
                          1.41421356237f, 2.2360679775f, 2.82842712475f,
                          2.2360679775f, 2.0f, 2.2360679775f, 2.82842712475f};

  float acc = 0.0f;
  for (int e = tid; e < PLANE; e += NTHREADS) {
    const int h = e >> 7;
    const int w = e & (TILE_W - 1);
    const float center = tile[e];
#pragma unroll
    for (int k = 0; k < NOFF; ++k) {
      const int hh = h - oi[k];
      const int ww = w - oj[k];
      const bool ok = (hh >= 0) & (ww >= 0) & (ww < TILE_W);
      const int hc = hh < 0 ? 0 : hh;
      const int wc = ww < 0 ? 0 : (ww >= TILE_W ? TILE_W - 1 : ww);
      const float nb = tile[hc * TILE_W + wc];
      const float d  = ok ? (center - nb) : 0.0f;
      const int aj = oj[k] < 0 ? -oj[k] : oj[k];
      const float scale = 1.0f /
          ((float)(TILE_H - oi[k]) * (float)(TILE_W - aj) * rd[k]);
      acc = fmaf(d * d, scale, acc);
    }
  }

  // ---- Block reduction: WMMA wave all-reduce, then 8 wave partials --------
  const float wsum = wave_allreduce(acc);   // all threads execute (EXEC all 1s)

  __shared__ float wpart[NTHREADS / 32];
  if ((tid & 31) == 0) wpart[tid >> 5] = wsum;
  __syncthreads();
  if (tid == 0) {
    float s = 0.0f;
#pragma unroll
    for (int i = 0; i < NTHREADS / 32; ++i) s += wpart[i];
    partials[plane] = s;
  }
}

// ---------------- Kernel 2: deterministic final reduction ---------------------
__global__ __launch_bounds__(NTHREADS)
void spatial_loss_final(const float* __restrict__ partials, int n,
                        float final_scale, float* __restrict__ out)
{
  const int tid = threadIdx.x;
  float s = 0.0f;
  for (int i = tid; i < n; i += NTHREADS) s += partials[i];   // fixed order

  const float wsum = wave_allreduce(s);

  __shared__ float wpart[NTHREADS / 32];
  if ((tid & 31) == 0) wpart[tid >> 5] = wsum;
  __syncthreads();
  if (tid == 0) {
    float t = 0.0f;
#pragma unroll
    for (int i = 0; i < NTHREADS / 32; ++i) t += wpart[i];
    out[0] = t * final_scale;   // factor 2 (offset symmetry) / (T*C)
  }
}

extern "C" void kernel_launch(void* const* d_in, const int* in_sizes, int n_in,
                              void* d_out, int out_size, void* d_ws, size_t ws_size,
                              hipStream_t stream)
{
  const float* x = (const float*)d_in[0];
  const int total   = in_sizes[0];          // T*C*H*W = 33,554,432
  const int nplanes = total / PLANE;        // T*C = 2048
  float* partials = (float*)d_ws;           // 2048 floats << ws_size

  spatial_loss_plane<<<nplanes, NTHREADS, 0, stream>>>(x, partials);
  spatial_loss_final<<<1, NTHREADS, 0, stream>>>(
      partials, nplanes, 2.0f / (float)nplanes, (float*)d_out);
}